// LSTM_29703993819793
// MI455X (gfx1250) — compile-verified
//
#include <hip/hip_runtime.h>
#include <hip/hip_bf16.h>

// Problem constants (reference: B=512, T=1024, I=64, H=10)
static constexpr int BATCH = 512;
static constexpr int SEQ   = 1024;
static constexpr int IN    = 64;
static constexpr int HID   = 10;     // 4H = 40 gates, padded to 48 in xg
static constexpr int GP    = 48;     // padded gate stride (3 x 16 WMMA N-tiles)

typedef __attribute__((ext_vector_type(2))) float v2f;
typedef __attribute__((ext_vector_type(8))) float v8f;

// ---------------------------------------------------------------------------
// Kernel 1: xg = x @ Wih1^T + (bih1 + bhh1), fp32 WMMA 16x16x4.
//   x  : [B*T, 64] row-major (row r = b*T + t)
//   out: [t][b][48]  (scan-friendly layout; cols 40..47 are zero-weight pad)
// One wave handles one 16-row M tile x 3 N tiles; K=64 -> 16 wmma k-steps,
// A fragment reused across the 3 N tiles.
// ---------------------------------------------------------------------------
__global__ __launch_bounds__(256)
void xg_gemm_f32(const float* __restrict__ x,
                 const float* __restrict__ Wih,   // [40][64]
                 const float* __restrict__ bih,   // [40]
                 const float* __restrict__ bhh,   // [40]
                 float* __restrict__ xg)          // [T][B][48]
{
    __shared__ float sW[GP * IN];                 // 12 KB, rows 40..47 zeroed
    for (int i = threadIdx.x; i < GP * IN; i += blockDim.x)
        sW[i] = (i < 40 * IN) ? Wih[i] : 0.0f;
    __syncthreads();

    const int lane  = threadIdx.x & 31;
    const int wave  = threadIdx.x >> 5;
    const int tile  = blockIdx.x * 8 + wave;      // M tile (16 rows of B*T)
    const int row0  = tile << 4;
    const int ncol  = lane & 15;                  // N within tile / A row
    const int khalf = (lane >> 4) << 1;           // 0 or 2 (A/B K split)
    const int mbase = (lane >> 4) << 3;           // 0 or 8 (D row split)

    // C init = per-column bias (broadcast down the 8 accumulator VGPRs)
    const float b0 = bih[ncol]      + bhh[ncol];
    const float b1 = bih[16 + ncol] + bhh[16 + ncol];
    const float b2 = (ncol < 8) ? (bih[32 + ncol] + bhh[32 + ncol]) : 0.0f;
    v8f c0, c1, c2;
#pragma unroll
    for (int v = 0; v < 8; ++v) { c0[v] = b0; c1[v] = b1; c2[v] = b2; }

    const float* arow = x  + (size_t)(row0 + ncol) * IN + khalf;
    const float* w0   = sW + (  0 + ncol) * IN + khalf;
    const float* w1   = sW + ( 16 + ncol) * IN + khalf;
    const float* w2   = sW + ( 32 + ncol) * IN + khalf;

#pragma unroll
    for (int ks = 0; ks < 16; ++ks) {
        const int k = ks * 4;
        const v2f a  = *(const v2f*)(arow + k);
        const v2f f0 = *(const v2f*)(w0 + k);
        const v2f f1 = *(const v2f*)(w1 + k);
        const v2f f2 = *(const v2f*)(w2 + k);
        c0 = __builtin_amdgcn_wmma_f32_16x16x4_f32(false, a, false, f0,
                                                   (short)0, c0, false, false);
        c1 = __builtin_amdgcn_wmma_f32_16x16x4_f32(false, a, false, f1,
                                                   (short)0, c1, false, false);
        c2 = __builtin_amdgcn_wmma_f32_16x16x4_f32(false, a, false, f2,
                                                   (short)0, c2, false, false);
    }

    // D layout: VGPR v -> M = v (lanes 0-15) / v+8 (lanes 16-31), N = lane%16
#pragma unroll
    for (int v = 0; v < 8; ++v) {
        const int r  = row0 + mbase + v;          // r = b*T + t
        const int bb = r >> 10;
        const int tt = r & (SEQ - 1);
        float* orow = xg + ((size_t)tt * BATCH + (size_t)bb) * GP;
        orow[ncol]      = c0[v];
        orow[16 + ncol] = c1[v];
        orow[32 + ncol] = c2[v];
    }
}

// ---------------------------------------------------------------------------
// Fast activations (hardware v_exp path)
// ---------------------------------------------------------------------------
__device__ __forceinline__ float sigm(float v)  { return 1.0f / (1.0f + __expf(-v)); }
__device__ __forceinline__ float tanhx(float v) { return 2.0f / (1.0f + __expf(-2.0f * v)) - 1.0f; }

// ---------------------------------------------------------------------------
// Kernel 2: layer-1 recurrence. One thread per batch element; Whh1 in LDS
// (wave-uniform broadcast reads); h,c in registers; writes y1 [t][b][10]
// and the final (h,c) that seed layer 2.
// ---------------------------------------------------------------------------
__global__ __launch_bounds__(32)
void lstm_scan1(const float* __restrict__ xg,    // [T][B][48]
                const float* __restrict__ Whh,   // [40][10]
                const float* __restrict__ h0,    // [B][10]
                const float* __restrict__ c0,    // [B][10]
                float* __restrict__ y1,          // [T][B][10]
                float* __restrict__ hT,          // [B][10]
                float* __restrict__ cT)          // [B][10]
{
    __shared__ float sW[40 * HID];
    for (int i = threadIdx.x; i < 40 * HID; i += blockDim.x) sW[i] = Whh[i];
    __syncthreads();

    const int b = blockIdx.x * blockDim.x + threadIdx.x;
    float h[HID], c[HID];
#pragma unroll
    for (int j = 0; j < HID; ++j) { h[j] = h0[b * HID + j]; c[j] = c0[b * HID + j]; }

    for (int t = 0; t < SEQ; ++t) {
        const float* xr = xg + ((size_t)t * BATCH + b) * GP;
        if (t + 1 < SEQ)
            __builtin_prefetch(xg + ((size_t)(t + 1) * BATCH + b) * GP, 0, 0);

        float g[40];
#pragma unroll
        for (int q = 0; q < 10; ++q) {
            const float4 v = *(const float4*)(xr + 4 * q);
            g[4 * q] = v.x; g[4 * q + 1] = v.y; g[4 * q + 2] = v.z; g[4 * q + 3] = v.w;
        }
#pragma unroll
        for (int gi = 0; gi < 40; ++gi) {
            float a = g[gi];
#pragma unroll
            for (int j = 0; j < HID; ++j) a += sW[gi * HID + j] * h[j];
            g[gi] = a;
        }
        float* yr = y1 + ((size_t)t * BATCH + b) * HID;
#pragma unroll
        for (int j = 0; j < HID; ++j) {
            const float ig = sigm(g[j]);
            const float fg = sigm(g[10 + j]);
            const float gg = tanhx(g[20 + j]);
            const float og = sigm(g[30 + j]);
            c[j] = fg * c[j] + ig * gg;
            h[j] = og * tanhx(c[j]);
            yr[j] = h[j];
        }
    }
#pragma unroll
    for (int j = 0; j < HID; ++j) { hT[b * HID + j] = h[j]; cT[b * HID + j] = c[j]; }
}

// ---------------------------------------------------------------------------
// Kernel 3: layer-2 recurrence fused with fc1 and fc2.
// gates = (Wih2 @ y1_t + bih2 + bhh2) + Whh2 @ h ; out accumulates
// (fc1_w . h_t + fc1_b) * fc2_w[t] so y2 is never stored.
// ---------------------------------------------------------------------------
__global__ __launch_bounds__(32)
void lstm_scan2(const float* __restrict__ y1,    // [T][B][10]
                const float* __restrict__ Wih,   // [40][10]
                const float* __restrict__ Whh,   // [40][10]
                const float* __restrict__ bih,   // [40]
                const float* __restrict__ bhh,   // [40]
                const float* __restrict__ h0,    // [B][10] = layer-1 final h
                const float* __restrict__ c0,    // [B][10] = layer-1 final c
                const float* __restrict__ fc1w,  // [10]
                const float* __restrict__ fc1b,  // [1]
                const float* __restrict__ fc2w,  // [1024]
                const float* __restrict__ fc2b,  // [1]
                float* __restrict__ out)         // [B]
{
    __shared__ float sWi[40 * HID], sWh[40 * HID], sB[40], sF[HID];
    for (int i = threadIdx.x; i < 40 * HID; i += blockDim.x) {
        sWi[i] = Wih[i]; sWh[i] = Whh[i];
    }
    for (int i = threadIdx.x; i < 40; i += blockDim.x) sB[i] = bih[i] + bhh[i];
    for (int i = threadIdx.x; i < HID; i += blockDim.x) sF[i] = fc1w[i];
    __syncthreads();

    const int b = blockIdx.x * blockDim.x + threadIdx.x;
    float h[HID], c[HID];
#pragma unroll
    for (int j = 0; j < HID; ++j) { h[j] = h0[b * HID + j]; c[j] = c0[b * HID + j]; }

    const float f1b = fc1b[0];
    float acc = 0.0f;

    for (int t = 0; t < SEQ; ++t) {
        const float* yr = y1 + ((size_t)t * BATCH + b) * HID;
        if (t + 1 < SEQ)
            __builtin_prefetch(y1 + ((size_t)(t + 1) * BATCH + b) * HID, 0, 0);
        float yv[HID];
#pragma unroll
        for (int j = 0; j < HID; ++j) yv[j] = yr[j];

        float g[40];
#pragma unroll
        for (int gi = 0; gi < 40; ++gi) {
            float a = sB[gi];
#pragma unroll
            for (int j = 0; j < HID; ++j)
                a += sWi[gi * HID + j] * yv[j] + sWh[gi * HID + j] * h[j];
            g[gi] = a;
        }
#pragma unroll
        for (int j = 0; j < HID; ++j) {
            const float ig = sigm(g[j]);
            const float fg = sigm(g[10 + j]);
            const float gg = tanhx(g[20 + j]);
            const float og = sigm(g[30 + j]);
            c[j] = fg * c[j] + ig * gg;
            h[j] = og * tanhx(c[j]);
        }
        float p = f1b;
#pragma unroll
        for (int j = 0; j < HID; ++j) p += sF[j] * h[j];
        acc += p * fc2w[t];                       // wave-uniform -> scalar load
    }
    out[b] = acc + fc2b[0];
}

// ---------------------------------------------------------------------------
extern "C" void kernel_launch(void* const* d_in, const int* in_sizes, int n_in,
                              void* d_out, int out_size, void* d_ws, size_t ws_size,
                              hipStream_t stream) {
    const float* x    = (const float*)d_in[0];
    const float* h0   = (const float*)d_in[1];
    const float* c0   = (const float*)d_in[2];
    const float* Wih1 = (const float*)d_in[3];
    const float* Whh1 = (const float*)d_in[4];
    const float* bih1 = (const float*)d_in[5];
    const float* bhh1 = (const float*)d_in[6];
    const float* Wih2 = (const float*)d_in[7];
    const float* Whh2 = (const float*)d_in[8];
    const float* bih2 = (const float*)d_in[9];
    const float* bhh2 = (const float*)d_in[10];
    const float* fc1w = (const float*)d_in[11];
    const float* fc1b = (const float*)d_in[12];
    const float* fc2w = (const float*)d_in[13];
    const float* fc2b = (const float*)d_in[14];
    float* out = (float*)d_out;

    // Workspace layout (floats): xg [T][B][48] | y1 [T][B][10] | hT | cT
    float* ws = (float*)d_ws;
    float* xg = ws;
    float* y1 = xg + (size_t)SEQ * BATCH * GP;     // 25,165,824 floats
    float* hT = y1 + (size_t)SEQ * BATCH * HID;    //  5,242,880 floats
    float* cT = hT + (size_t)BATCH * HID;

    // 1) Input projection GEMM: 32768 M-tiles, 8 waves/block -> 4096 blocks
    xg_gemm_f32<<<(BATCH * SEQ / 16) / 8, 256, 0, stream>>>(x, Wih1, bih1, bhh1, xg);
    // 2) Layer-1 scan: 512 chains, 16 single-wave blocks across WGPs
    lstm_scan1<<<BATCH / 32, 32, 0, stream>>>(xg, Whh1, h0, c0, y1, hT, cT);
    // 3) Layer-2 scan + fc1 + fc2 fused
    lstm_scan2<<<BATCH / 32, 32, 0, stream>>>(y1, Wih2, Whh2, bih2, bhh2,
                                              hT, cT, fc1w, fc1b, fc2w, fc2b, out);
}